// ProjectorSpanProto_53060025975428
// MI455X (gfx1250) — compile-verified
//
#include <hip/hip_runtime.h>
#include <math.h>

// Problem constants (reference: T,H,S = 131072, 768, 16384; MAX_LEN = 32)
#define H_DIM 768
#define S_SPN 16384
#define H4    (H_DIM / 4)   // 192 float4 per H-row

typedef float v2f __attribute__((ext_vector_type(2)));
typedef float v8f __attribute__((ext_vector_type(8)));

// -------- gfx1250 async global->LDS path (guarded; falls back cleanly) -----
#if defined(__gfx1250__) && __has_builtin(__builtin_amdgcn_global_load_async_to_lds_b128)
#define HAVE_ASYNC_LDS 1
// Probe-derived parameter type: plain pointer to 128-bit int vector.
typedef int v4i_vs __attribute__((vector_size(4 * sizeof(int))));
#define ASYNC_COPY16(dst_lds, src_glb)                                        \
  __builtin_amdgcn_global_load_async_to_lds_b128(                             \
      (v4i_vs*)(src_glb), (v4i_vs*)(dst_lds), 0, 0)
#if __has_builtin(__builtin_amdgcn_s_wait_asynccnt)
#define WAIT_ASYNC(n) __builtin_amdgcn_s_wait_asynccnt(n)
#else
#define WAIT_ASYNC(n) asm volatile("s_wait_asynccnt %0" ::"n"(n) : "memory")
#endif
#else
#define HAVE_ASYNC_LDS 0
#endif

// ---------------------------------------------------------------------------
// Kernel 1: build feat = [h_start | h_end | h_mean]  (S x 3H), fp32.
// One 192-thread block (6 waves) per span; thread t owns float4 column t.
// Pure HBM streaming: ~len*3KB per span, coalesced.
// ---------------------------------------------------------------------------
__global__ __launch_bounds__(H4)
void span_feat_kernel(const float* __restrict__ tok,
                      const long long* __restrict__ spans,
                      float* __restrict__ feat) {
  const int s = blockIdx.x;
  const int t = threadIdx.x;                  // 0..191
  const long long st = spans[2 * s + 0];
  const long long en = spans[2 * s + 1];
  const int len = (int)(en - st);

  float4 acc = make_float4(0.f, 0.f, 0.f, 0.f);
  float4 hs  = make_float4(0.f, 0.f, 0.f, 0.f);
  float4 he  = make_float4(0.f, 0.f, 0.f, 0.f);
  for (int r = 0; r < len; ++r) {
    float4 v = ((const float4*)(tok + (size_t)(st + r) * H_DIM))[t];
    if (r == 0) hs = v;
    he = v;                                   // keeps last row
    acc.x += v.x; acc.y += v.y; acc.z += v.z; acc.w += v.w;
  }
  const float inv = 1.0f / (float)len;
  float4 hm = make_float4(acc.x * inv, acc.y * inv, acc.z * inv, acc.w * inv);

  float4* frow = (float4*)(feat + (size_t)s * (3 * H_DIM));
  frow[t]          = hs;
  frow[t + H4]     = he;
  frow[t + 2 * H4] = hm;
}

// ---------------------------------------------------------------------------
// Kernel 2/3: C = gelu_exact(A(MxK) * B(KxN) + bias), fp32 WMMA 16x16x4.
// Block tile 128x64, 8 waves, each wave a 32x32 tile (2x2 v8f accumulators).
// Double-buffered LDS; async global->LDS (ASYNCcnt) when available.
// Pads chosen so every 16B chunk is 16B-aligned and frag loads are
// bank-conflict-free (A stride 36 dw, B stride 72 dw; half-waves 16 banks apart).
// ---------------------------------------------------------------------------
#define BM 128
#define BN 64
#define BK 32
#define APAD 4
#define BPAD 8

__global__ __launch_bounds__(256)
void gemm_bias_gelu_kernel(const float* __restrict__ A,
                           const float* __restrict__ B,
                           const float* __restrict__ bias,
                           float* __restrict__ C,
                           int M, int N, int K) {
  __shared__ float As[2][BM][BK + APAD];
  __shared__ float Bs[2][BK][BN + BPAD];

  const int tid  = threadIdx.x;
  const int wave = tid >> 5;
  const int lane = tid & 31;
  const int bm = blockIdx.y * BM;
  const int bn = blockIdx.x * BN;
  const int wm = (wave >> 1) * 32;   // wave M offset within block tile (0..96)
  const int wn = (wave & 1) * 32;    // wave N offset within block tile (0 or 32)

  const int l15   = lane & 15;
  const int khalf = (lane >> 4) << 1;  // 0 or 2 : K half per A/B frag layout
  const int mhalf = (lane >> 4) << 3;  // 0 or 8 : M half for C/D layout

  // Per-thread staging coordinates (16B granules)
  const int arow0 = tid >> 1;                 // A: 1 row/thread-pair, 8 granules/row
  const int acol0 = (tid & 1) << 4;           // 0 or 16 (four float4 each)
  const int brow0 = tid >> 4;                 // B rows 0..15 (+16 second pass)
  const int bcol0 = (tid & 15) << 2;          // 16 float4 per row

  v8f acc[2][2];
#pragma unroll
  for (int im = 0; im < 2; ++im)
#pragma unroll
    for (int in = 0; in < 2; ++in)
#pragma unroll
      for (int r = 0; r < 8; ++r) acc[im][in][r] = 0.0f;

  // ---- tile staging: 4x A-b128 + 2x B-b128 per thread = 6 async ops ----
#if HAVE_ASYNC_LDS
#define STAGE16(dst, src) ASYNC_COPY16((dst), (src))
#else
#define STAGE16(dst, src) (*(float4*)(dst) = *(const float4*)(src))
#endif

#define STAGE_TILE(buf, kbase)                                                 \
  do {                                                                         \
    const float* a0 = A + (size_t)(bm + arow0) * K + ((kbase) + acol0);        \
    const float* b0 = B + (size_t)((kbase) + brow0) * N + (bn + bcol0);        \
    const float* b1p = B + (size_t)((kbase) + brow0 + 16) * N + (bn + bcol0);  \
    STAGE16(&As[buf][arow0][acol0 + 0],  a0 + 0);                              \
    STAGE16(&As[buf][arow0][acol0 + 4],  a0 + 4);                              \
    STAGE16(&As[buf][arow0][acol0 + 8],  a0 + 8);                              \
    STAGE16(&As[buf][arow0][acol0 + 12], a0 + 12);                             \
    STAGE16(&Bs[buf][brow0][bcol0],      b0);                                  \
    STAGE16(&Bs[buf][brow0 + 16][bcol0], b1p);                                 \
  } while (0)

  // Prologue: stage tile 0 into buffer 0
  STAGE_TILE(0, 0);

  int buf = 0;
  for (int k0 = 0; k0 < K; k0 += BK, buf ^= 1) {
    const bool has_next = (k0 + BK) < K;
    if (has_next) STAGE_TILE(buf ^ 1, k0 + BK);
#if HAVE_ASYNC_LDS
    if (has_next) { WAIT_ASYNC(6); } else { WAIT_ASYNC(0); }
#endif
    __syncthreads();   // current tile resident for all waves

    // ---- 8 K-steps of 4, 4 WMMAs each ----
#pragma unroll
    for (int kk = 0; kk < BK; kk += 4) {
      v2f afrag[2], bfrag[2];
#pragma unroll
      for (int im = 0; im < 2; ++im) {
        int m = wm + im * 16 + l15;
        afrag[im][0] = As[buf][m][kk + khalf + 0];
        afrag[im][1] = As[buf][m][kk + khalf + 1];
      }
#pragma unroll
      for (int in = 0; in < 2; ++in) {
        int n = wn + in * 16 + l15;
        bfrag[in][0] = Bs[buf][kk + khalf + 0][n];
        bfrag[in][1] = Bs[buf][kk + khalf + 1][n];
      }
#pragma unroll
      for (int im = 0; im < 2; ++im)
#pragma unroll
        for (int in = 0; in < 2; ++in)
          acc[im][in] = __builtin_amdgcn_wmma_f32_16x16x4_f32(
              false, afrag[im], false, bfrag[in],
              (short)0, acc[im][in], false, false);
    }
    __syncthreads();   // all reads of `buf` done before it is restaged
  }

  // ---- epilogue: bias + exact GELU + store ----
#pragma unroll
  for (int im = 0; im < 2; ++im) {
#pragma unroll
    for (int in = 0; in < 2; ++in) {
      int n = bn + wn + in * 16 + l15;
      float bv = bias[n];
#pragma unroll
      for (int r = 0; r < 8; ++r) {
        int m = bm + wm + im * 16 + mhalf + r;
        float x = acc[im][in][r] + bv;
        float g = 0.5f * x * (1.0f + erff(x * 0.70710678118654752f));
        C[(size_t)m * N + n] = g;
      }
    }
  }
}

// ---------------------------------------------------------------------------
// Launch: feat -> gemm1(gelu) -> gemm2(gelu)
// ---------------------------------------------------------------------------
extern "C" void kernel_launch(void* const* d_in, const int* in_sizes, int n_in,
                              void* d_out, int out_size, void* d_ws, size_t ws_size,
                              hipStream_t stream) {
  const float*     tok   = (const float*)d_in[0];
  const long long* spans = (const long long*)d_in[1];
  const float*     W1    = (const float*)d_in[2];
  const float*     b1    = (const float*)d_in[3];
  const float*     W2    = (const float*)d_in[4];
  const float*     b2    = (const float*)d_in[5];
  float*           out   = (float*)d_out;

  float* feat = (float*)d_ws;                               // S x 3H  (151 MB)
  float* z    = feat + (size_t)S_SPN * 3 * H_DIM;           // S x H   (50 MB)

  // 1) span features
  span_feat_kernel<<<S_SPN, H4, 0, stream>>>(tok, spans, feat);

  // 2) z = gelu(feat @ W1 + b1):  M=S, K=3H, N=H
  dim3 grid(H_DIM / BN, S_SPN / BM);   // (12, 128)
  gemm_bias_gelu_kernel<<<grid, 256, 0, stream>>>(feat, W1, b1, z,
                                                  S_SPN, H_DIM, 3 * H_DIM);

  // 3) out = gelu(z @ W2 + b2):  M=S, K=H, N=H
  gemm_bias_gelu_kernel<<<grid, 256, 0, stream>>>(z, W2, b2, out,
                                                  S_SPN, H_DIM, H_DIM);
}